// NonMaximumSuppression_9577777070544
// MI455X (gfx1250) — compile-verified
//
#include <hip/hip_runtime.h>
#include <cstdint>

// NMS (3x3 masked max suppression) over (N=16, H=512, W=512, C=8) fp32.
// HBM-bound: ~268 MB traffic -> ~11.5 us floor at 23.3 TB/s.
// gfx1250 path: global_load_async_to_lds_b128 staging (ASYNCcnt) + b128 LDS
// reads + nontemporal b128 stores; wave32, 8 waves/block, 10.6 KB LDS/WG.

typedef float v4f __attribute__((ext_vector_type(4)));

namespace {
constexpr int Himg = 512;
constexpr int Wimg = 512;
constexpr int Cch  = 8;        // 8 floats = 32 B per pixel
constexpr int TW   = 32;       // tile width  (pixels)
constexpr int TH   = 8;        // tile height (rows)
constexpr int LW   = TW + 2;   // 34 pixels incl. halo
constexpr int LH   = TH + 2;   // 10 rows incl. halo
constexpr int THREADS = 256;   // 8 wave32
constexpr int CHUNKS  = LH * LW * 2;  // 16B chunks to stage = 680
constexpr int ITERS   = (CHUNKS + THREADS - 1) / THREADS;  // 3
}

__global__ __launch_bounds__(THREADS)
void nms_async_tile_kernel(const float* __restrict__ x,
                           const unsigned char* __restrict__ mask,
                           float* __restrict__ out) {
  __shared__ __align__(16) float tile[LH * LW * Cch];   // 10,880 B

  const int tid = threadIdx.x;
  const int tx0 = blockIdx.x * TW;
  const int ty0 = blockIdx.y * TH;
  const size_t img_base = (size_t)blockIdx.z * ((size_t)Himg * Wimg * Cch);

  // Low 32 bits of a generic pointer to LDS == byte offset in group segment.
  const uint32_t lds_base = (uint32_t)(uintptr_t)&tile[0];

  // ---- Stage (TH+2)x(TW+2) pixel tile via async global->LDS DMA (b128) ----
  // All 32 lanes of every wave issue: OOB coords are clamped to valid pixels
  // (those LDS slots are never read by compute), OOB chunk ids clamp to the
  // last chunk (duplicate identical writes). EXEC stays all-ones.
#pragma unroll
  for (int it = 0; it < ITERS; ++it) {
    int k = it * THREADS + tid;
    k = k < CHUNKS ? k : (CHUNKS - 1);
    const int r      = k / (2 * LW);        // halo row   0..9
    const int j      = k - r * (2 * LW);    // chunk in row 0..67
    const int p      = j >> 1;              // pixel in row 0..33
    const int half16 = (j & 1) << 4;        // 0 or 16 bytes
    int gy = ty0 - 1 + r;
    int gx = tx0 - 1 + p;
    gy = min(max(gy, 0), Himg - 1);
    gx = min(max(gx, 0), Wimg - 1);
    const uint32_t goff =
        (uint32_t)((img_base + ((size_t)gy * Wimg + gx) * Cch) * sizeof(float))
        + (uint32_t)half16;
    const uint32_t loff =
        lds_base + (uint32_t)((r * LW + p) * Cch * sizeof(float)) + (uint32_t)half16;
    asm volatile("global_load_async_to_lds_b128 %0, %1, %2"
                 :
                 : "v"(loff), "v"(goff), "s"(x)
                 : "memory");
  }

  // Drain this wave's async DMA, then sync the workgroup (cross-wave LDS use).
  asm volatile("s_wait_asynccnt 0" ::: "memory");
  __syncthreads();

  // 3x3 mask (jnp bool -> 1 byte/elem), uniform across the grid.
  bool mk[9];
#pragma unroll
  for (int i = 0; i < 9; ++i) mk[i] = mask[i] != 0;

  // One pixel (8 channels = 2x v4f) per thread.
  const int px = tid & (TW - 1);
  const int py = tid >> 5;
  const int gx = tx0 + px;
  const int gy = ty0 + py;

  const float* cptr = &tile[((py + 1) * LW + (px + 1)) * Cch];
  const v4f c0 = *(const v4f*)(cptr);
  const v4f c1 = *(const v4f*)(cptr + 4);

  const float ninf = -__builtin_huge_valf();
  v4f m0 = {ninf, ninf, ninf, ninf};
  v4f m1 = m0;

#pragma unroll
  for (int di = 0; di < 3; ++di) {
#pragma unroll
    for (int dj = 0; dj < 3; ++dj) {
      const int ny = gy + di - 1;
      const int nx = gx + dj - 1;
      // Skipping == treating the neighbor as -inf (mask-off or border pad).
      if (mk[di * 3 + dj] && ny >= 0 && ny < Himg && nx >= 0 && nx < Wimg) {
        const float* q = &tile[((py + di) * LW + (px + dj)) * Cch];
        const v4f a = *(const v4f*)(q);
        const v4f b = *(const v4f*)(q + 4);
        m0 = __builtin_elementwise_max(m0, a);
        m1 = __builtin_elementwise_max(m1, b);
      }
    }
  }

  v4f o0, o1;
  o0.x = c0.x > m0.x ? c0.x : 0.0f;
  o0.y = c0.y > m0.y ? c0.y : 0.0f;
  o0.z = c0.z > m0.z ? c0.z : 0.0f;
  o0.w = c0.w > m0.w ? c0.w : 0.0f;
  o1.x = c1.x > m1.x ? c1.x : 0.0f;
  o1.y = c1.y > m1.y ? c1.y : 0.0f;
  o1.z = c1.z > m1.z ? c1.z : 0.0f;
  o1.w = c1.w > m1.w ? c1.w : 0.0f;

  // Streaming output: nontemporal so it doesn't evict reusable input from L2.
  float* op = out + img_base + ((size_t)gy * Wimg + gx) * Cch;
  __builtin_nontemporal_store(o0, (v4f*)op);
  __builtin_nontemporal_store(o1, (v4f*)(op + 4));
}

extern "C" void kernel_launch(void* const* d_in, const int* in_sizes, int n_in,
                              void* d_out, int out_size, void* d_ws, size_t ws_size,
                              hipStream_t stream) {
  const float* x = (const float*)d_in[0];
  const unsigned char* mask = (const unsigned char*)d_in[1];
  float* out = (float*)d_out;

  const int N = in_sizes[0] / (Himg * Wimg * Cch);   // = 16
  dim3 grid(Wimg / TW, Himg / TH, N);                // (16, 64, 16)
  nms_async_tile_kernel<<<grid, THREADS, 0, stream>>>(x, mask, out);
}